// Attention_Weight_Sum_61873298866222
// MI455X (gfx1250) — compile-verified
//
#include <hip/hip_runtime.h>
#include <hip/hip_bf16.h>
#include <cmath>

#define HEADS 4
#define FEATS 32
#define HF (HEADS * FEATS)   // 128
#define NEG_SLOPE 0.2f
#define EPSV 1e-16f

typedef __attribute__((ext_vector_type(2))) float v2f;
typedef __attribute__((ext_vector_type(8))) float v8f;

// ---- order-preserving float<->uint mapping for atomic max ----
__device__ __forceinline__ unsigned fmap(float x) {
    unsigned b = __float_as_uint(x);
    return (b & 0x80000000u) ? ~b : (b | 0x80000000u);
}
__device__ __forceinline__ float funmap(unsigned u) {
    unsigned b = (u & 0x80000000u) ? (u & 0x7FFFFFFFu) : ~u;
    return __uint_as_float(b);
}

// ---------------------------------------------------------------------------
// K0: init node sums + global max cell (must run every launch: ws not reset)
// ---------------------------------------------------------------------------
__global__ void k_init(float* __restrict__ node_sums, unsigned* __restrict__ max_bits, int nh) {
    int i = blockIdx.x * blockDim.x + threadIdx.x;
    if (i < nh) node_sums[i] = 0.0f;
    if (i == 0) *max_bits = 0u;   // 0 <= fmap(x) for all finite x
}

// ---------------------------------------------------------------------------
// K1: per-node scores via V_WMMA_F32_16X16X4_F32.
// One wave computes a 16-node tile. A = [16 x 128] node features, accumulated
// over 32 K-chunks of 4. B[k][n] is block-structured: col n<4 -> source
// scorer of head n, col 4..7 -> target scorer of head n-4, zero elsewhere.
// Selection is branch-free: per-head-block {0,1} scalars applied with FMA.
// C[m][n<4] = scores_src[node m, head n]; C[m][4..7] = scores_trg.
// ---------------------------------------------------------------------------
__global__ void __launch_bounds__(256)
k_scores_wmma(const float* __restrict__ feat,
              const float* __restrict__ s_src,   // 128 floats (h*32+f)
              const float* __restrict__ s_trg,   // 128 floats
              float* __restrict__ scores_src,    // [N,H]
              float* __restrict__ scores_trg,    // [N,H]
              int N, int ntiles) {
    int wave = (blockIdx.x * blockDim.x + threadIdx.x) >> 5;
    if (wave >= ntiles) return;                    // wave-uniform exit, EXEC stays full
    int lane = threadIdx.x & 31;
    int m    = lane & 15;                          // row (A) / col (B,C)
    int kofs = (lane >> 4) << 1;                   // 0 or 2 within the K=4 chunk

    int node_base = wave * 16;
    int node_ld   = min(node_base + m, N - 1);     // clamp for safety (N%16==0 here)
    const float* arow = feat + (long)node_ld * HF;

    v8f c = {};
    #pragma unroll
    for (int h = 0; h < HEADS; ++h) {
        // branch-free B selection, hoisted per head block
        float sel_s = (m == h)         ? 1.0f : 0.0f;
        float sel_t = (m == h + HEADS) ? 1.0f : 0.0f;
        #pragma unroll
        for (int j = 0; j < FEATS / 4; ++j) {      // 8 K-chunks of 4 per head
            int k0 = h * FEATS + j * 4 + kofs;
            v2f a;
            a.x = arow[k0];
            a.y = arow[k0 + 1];
            float2 vs = *(const float2*)(s_src + k0);
            float2 vt = *(const float2*)(s_trg + k0);
            v2f b;
            b.x = sel_s * vs.x + sel_t * vt.x;
            b.y = sel_s * vs.y + sel_t * vt.y;
            c = __builtin_amdgcn_wmma_f32_16x16x4_f32(false, a, false, b,
                                                      (short)0, c, false, false);
        }
    }

    // C layout: lane L holds column n = L&15; VGPR r holds row (L<16 ? r : 8+r)
    int n     = m;
    int mhalf = (lane >> 4) << 3;
    if (n < 8) {
        #pragma unroll
        for (int r = 0; r < 8; ++r) {
            int node = node_base + mhalf + r;
            if (node < N) {
                if (n < 4) scores_src[node * HEADS + n]       = c[r];
                else       scores_trg[node * HEADS + (n - 4)] = c[r];
            }
        }
    }
}

// ---------------------------------------------------------------------------
// K2: per-edge leaky-relu scores + global max reduction
// ---------------------------------------------------------------------------
__global__ void __launch_bounds__(256)
k_edge_leaky_max(const long long* __restrict__ eidx,  // [2,E]
                 const float* __restrict__ scores_src,
                 const float* __restrict__ scores_trg,
                 float* __restrict__ edge_sc,          // [E,H]
                 unsigned* __restrict__ max_bits,
                 int E) {
    __shared__ float smax[8];
    int e = blockIdx.x * blockDim.x + threadIdx.x;
    float lmax = -INFINITY;
    if (e < E) {
        int trg = (int)eidx[e];
        int src = (int)eidx[(long)E + e];
        float4 ss = *(const float4*)(scores_src + src * HEADS);
        float4 st = *(const float4*)(scores_trg + trg * HEADS);
        float4 s;
        s.x = ss.x + st.x; s.y = ss.y + st.y; s.z = ss.z + st.z; s.w = ss.w + st.w;
        s.x = s.x > 0.0f ? s.x : NEG_SLOPE * s.x;
        s.y = s.y > 0.0f ? s.y : NEG_SLOPE * s.y;
        s.z = s.z > 0.0f ? s.z : NEG_SLOPE * s.z;
        s.w = s.w > 0.0f ? s.w : NEG_SLOPE * s.w;
        *(float4*)(edge_sc + (long)e * HEADS) = s;
        lmax = fmaxf(fmaxf(s.x, s.y), fmaxf(s.z, s.w));
    }
    // wave32 reduce
    #pragma unroll
    for (int off = 16; off > 0; off >>= 1)
        lmax = fmaxf(lmax, __shfl_xor(lmax, off, 32));
    int lane = threadIdx.x & 31;
    int wid  = threadIdx.x >> 5;
    if (lane == 0) smax[wid] = lmax;
    __syncthreads();
    if (threadIdx.x == 0) {
        float bmax = smax[0];
        #pragma unroll
        for (int w = 1; w < 8; ++w) bmax = fmaxf(bmax, smax[w]);
        atomicMax(max_bits, fmap(bmax));
    }
}

// ---------------------------------------------------------------------------
// K3: exp(score - gmax), scatter-add into per-target-node sums
// ---------------------------------------------------------------------------
__global__ void __launch_bounds__(256)
k_exp_scatter(const long long* __restrict__ eidx,
              float* __restrict__ edge_sc,     // in: leaky; out: exp
              float* __restrict__ node_sums,   // [N,H]
              const unsigned* __restrict__ max_bits,
              int E) {
    int e = blockIdx.x * blockDim.x + threadIdx.x;
    if (e >= E) return;
    float gmax = funmap(*max_bits);
    int trg = (int)eidx[e];
    float4 s = *(const float4*)(edge_sc + (long)e * HEADS);
    float4 ex;
    ex.x = expf(s.x - gmax);
    ex.y = expf(s.y - gmax);
    ex.z = expf(s.z - gmax);
    ex.w = expf(s.w - gmax);
    *(float4*)(edge_sc + (long)e * HEADS) = ex;
    float* dst = node_sums + trg * HEADS;
    atomicAdd(dst + 0, ex.x);
    atomicAdd(dst + 1, ex.y);
    atomicAdd(dst + 2, ex.z);
    atomicAdd(dst + 3, ex.w);
}

// ---------------------------------------------------------------------------
// K4: attention = exp / (sum[trg] + eps)  ->  d_out[0 : E*H]
// ---------------------------------------------------------------------------
__global__ void __launch_bounds__(256)
k_attention(const long long* __restrict__ eidx,
            const float* __restrict__ edge_sc,
            const float* __restrict__ node_sums,
            float* __restrict__ out_att,
            int E) {
    int e = blockIdx.x * blockDim.x + threadIdx.x;
    if (e >= E) return;
    int trg = (int)eidx[e];
    float4 ex = *(const float4*)(edge_sc + (long)e * HEADS);
    float4 sm = *(const float4*)(node_sums + trg * HEADS);
    float4 a;
    a.x = ex.x / (sm.x + EPSV);
    a.y = ex.y / (sm.y + EPSV);
    a.z = ex.z / (sm.z + EPSV);
    a.w = ex.w / (sm.w + EPSV);
    *(float4*)(out_att + (long)e * HEADS) = a;
}

// ---------------------------------------------------------------------------
// K5: feature lift: out2[e,:,:] = feat[src[e],:,:]   (one wave per edge,
// each lane moves one float4 -> 512B contiguous per wave)
// ---------------------------------------------------------------------------
__global__ void __launch_bounds__(256)
k_gather_feat(const long long* __restrict__ eidx,
              const float* __restrict__ feat,
              float* __restrict__ out_feat,
              int E) {
    int gtid = blockIdx.x * blockDim.x + threadIdx.x;
    int e    = gtid >> 5;
    int lane = gtid & 31;
    if (e >= E) return;
    int src = (int)eidx[(long)E + e];
    const float4* s = (const float4*)(feat + (long)src * HF);
    float4*       d = (float4*)(out_feat + (long)e * HF);
    d[lane] = s[lane];
}

// ---------------------------------------------------------------------------
extern "C" void kernel_launch(void* const* d_in, const int* in_sizes, int n_in,
                              void* d_out, int out_size, void* d_ws, size_t ws_size,
                              hipStream_t stream) {
    const float*     feat  = (const float*)d_in[0];      // [N,H,F]
    const float*     s_src = (const float*)d_in[1];      // [1,H,F]
    const float*     s_trg = (const float*)d_in[2];      // [1,H,F]
    const long long* eidx  = (const long long*)d_in[3];  // [2,E] int64

    const int N = in_sizes[0] / HF;
    const int E = in_sizes[3] / 2;
    const int NH = N * HEADS;

    // workspace layout (floats); all section sizes are multiples of 4 -> 16B aligned
    float* ws          = (float*)d_ws;
    float* scores_src  = ws;                 // N*H
    float* scores_trg  = scores_src + NH;    // N*H
    float* node_sums   = scores_trg + NH;    // N*H
    float* edge_sc     = node_sums + NH;     // E*H
    unsigned* max_bits = (unsigned*)(edge_sc + (long)E * HEADS);

    float* out_att  = (float*)d_out;             // E*H
    float* out_feat = out_att + (long)E * HEADS; // E*H*F

    const int B = 256;

    // 0) init accumulators
    k_init<<<(NH + B - 1) / B, B, 0, stream>>>(node_sums, max_bits, NH);

    // 1) per-node scores via WMMA (one wave per 16-node tile)
    int ntiles = (N + 15) / 16;
    int waves_per_blk = B / 32;
    k_scores_wmma<<<(ntiles + waves_per_blk - 1) / waves_per_blk, B, 0, stream>>>(
        feat, s_src, s_trg, scores_src, scores_trg, N, ntiles);

    // 2) edge leaky scores + global max
    k_edge_leaky_max<<<(E + B - 1) / B, B, 0, stream>>>(
        eidx, scores_src, scores_trg, edge_sc, max_bits, E);

    // 3) exp + segment-sum scatter
    k_exp_scatter<<<(E + B - 1) / B, B, 0, stream>>>(
        eidx, edge_sc, node_sums, max_bits, E);

    // 4) normalize -> attention output
    k_attention<<<(E + B - 1) / B, B, 0, stream>>>(
        eidx, edge_sc, node_sums, out_att, E);

    // 5) feature gather -> second output (wave per edge)
    long gather_threads = (long)E * 32;
    k_gather_feat<<<(int)((gather_threads + B - 1) / B), B, 0, stream>>>(
        eidx, feat, out_feat, E);
}